// STGI_88725434400964
// MI455X (gfx1250) — compile-verified
//
#include <hip/hip_runtime.h>

typedef __attribute__((ext_vector_type(2))) float v2f;
typedef __attribute__((ext_vector_type(8))) float v8f;

#define T_DIM   16
#define N_NODES 10000
#define FIN     32
#define FH      64
#define E_EDGES 160000
#define M_MSGS  (E_EDGES + N_NODES)   // 170000 (edges + self-loops)

// workspace layout, in floats
#define OFF_DEG  0ll                      // N floats (deg, then overwritten by rsqrt -> dis)
#define OFF_NORM 16384ll                  // M floats
#define OFF_XW   262144ll                 // T*N*64 floats; reused as hw (T*N*32) for layer 2
#define OFF_H    (262144ll + 10240000ll)  // T*N*64 floats

// ---------------- norm computation ----------------

__global__ void k_init_deg(float* __restrict__ deg) {
  int n = blockIdx.x * blockDim.x + threadIdx.x;
  if (n < N_NODES) deg[n] = 1.0f;   // self-loop weight 1
}

__global__ void k_deg_accum(float* __restrict__ deg, const int* __restrict__ ei,
                            const float* __restrict__ ew) {
  int e = blockIdx.x * blockDim.x + threadIdx.x;   // grid sized exactly to E
  unsafeAtomicAdd(&deg[ei[E_EDGES + e]], ew[e]);   // dst-indexed segment sum
}

__global__ void k_dis(float* __restrict__ deg) {
  int n = blockIdx.x * blockDim.x + threadIdx.x;
  if (n < N_NODES) deg[n] = rsqrtf(deg[n]);        // deg >= 1 always
}

__global__ void k_norm(float* __restrict__ norm, const float* __restrict__ dis,
                       const int* __restrict__ ei, const float* __restrict__ ew) {
  int m = blockIdx.x * blockDim.x + threadIdx.x;
  if (m >= M_MSGS) return;
  if (m < E_EDGES) {
    norm[m] = dis[ei[m]] * ew[m] * dis[ei[E_EDGES + m]];
  } else {
    int n = m - E_EDGES;
    norm[m] = dis[n] * dis[n];                     // self-loop: 1/deg
  }
}

// ---------------- init / elementwise ----------------

__global__ void k_init_bias(float* __restrict__ buf, const float* __restrict__ bias,
                            int fmask, long long valid, long long total) {
  long long i = (long long)blockIdx.x * blockDim.x + threadIdx.x;
  if (i >= total) return;
  buf[i] = (i < valid) ? bias[(int)(i & fmask)] : 0.0f;  // tail (tuple scalar slot) -> 0
}

__global__ void k_relu4(float4* __restrict__ buf) {
  long long i = (long long)blockIdx.x * blockDim.x + threadIdx.x; // grid sized exactly
  float4 v = buf[i];
  v.x = fmaxf(v.x, 0.0f); v.y = fmaxf(v.y, 0.0f);
  v.z = fmaxf(v.z, 0.0f); v.w = fmaxf(v.w, 0.0f);
  buf[i] = v;
}

// ---------------- dense GEMM via V_WMMA_F32_16X16X4_F32 ----------------
// Out(R x FOUT) = A(R x K) @ W(K x FOUT), R = T*N = 160000 (divides 128 exactly).
// One wave computes a 16-row x FOUT stripe; 8 waves per 256-thread block.

template<int K, int FOUT>
__global__ void k_gemm_wmma(const float* __restrict__ A, const float* __restrict__ W,
                            float* __restrict__ Out) {
  const int lane = threadIdx.x & 31;
  const int wave = threadIdx.x >> 5;
  const int lid  = lane & 15;
  const int half = lane >> 4;
  const long long rowBase = (long long)blockIdx.x * 128 + (long long)wave * 16;

  constexpr int NT = FOUT / 16;
  v8f acc[NT];
#pragma unroll
  for (int ct = 0; ct < NT; ++ct) acc[ct] = (v8f){0,0,0,0,0,0,0,0};

#pragma unroll
  for (int k0 = 0; k0 < K; k0 += 4) {
    // A 16x4 f32 tile: lanes 0-15 rows M=0..15 hold K = k0+0,1 ; lanes 16-31 hold K = k0+2,3
    const float* ap = A + (rowBase + lid) * K + (k0 + 2 * half);
    v2f a; a.x = ap[0]; a.y = ap[1];
#pragma unroll
    for (int ct = 0; ct < NT; ++ct) {
      // B 4x16 tile: lane = column N, VGPR0/1 hold K = k0+2*half, k0+2*half+1
      const float* bp = W + (k0 + 2 * half) * FOUT + ct * 16 + lid;
      v2f b; b.x = bp[0]; b.y = bp[FOUT];
      acc[ct] = __builtin_amdgcn_wmma_f32_16x16x4_f32(
          /*neg_a=*/false, a, /*neg_b=*/false, b,
          /*c_mod=*/(short)0, acc[ct], /*reuse_a=*/false, /*reuse_b=*/false);
    }
  }

  // C/D layout: VGPR r -> row M = r + 8*half, col N = lid
#pragma unroll
  for (int ct = 0; ct < NT; ++ct) {
#pragma unroll
    for (int r = 0; r < 8; ++r) {
      Out[(rowBase + 8 * half + r) * FOUT + ct * 16 + lid] = acc[ct][r];
    }
  }
}

// ---------------- gather-scale-scatter (message passing) ----------------
// grid = (M*FOUT/256, T); feature index in low lane bits -> coalesced gathers,
// wave-uniform edge index / norm loads; hardware f32 atomics into dst rows.

template<int FLOG>
__global__ void k_scatter(const float* __restrict__ sf, float* __restrict__ df,
                          const int* __restrict__ ei, const float* __restrict__ norm) {
  constexpr int FOUT = 1 << FLOG;
  const int t = blockIdx.y;
  long long tid = (long long)blockIdx.x * blockDim.x + threadIdx.x; // exact coverage
  const int f = (int)(tid & (FOUT - 1));
  const int m = (int)(tid >> FLOG);
  int s, d;
  if (m < E_EDGES) { s = ei[m]; d = ei[E_EDGES + m]; }
  else             { s = d = m - E_EDGES; }
  const float nv = norm[m];
  const size_t base = (size_t)t * N_NODES * FOUT;
  const float val = sf[base + (size_t)s * FOUT + f] * nv;
  unsafeAtomicAdd(&df[base + (size_t)d * FOUT + f], val);
}

// ---------------- driver ----------------

extern "C" void kernel_launch(void* const* d_in, const int* in_sizes, int n_in,
                              void* d_out, int out_size, void* d_ws, size_t ws_size,
                              hipStream_t stream) {
  const float* x  = (const float*)d_in[0];
  const int*   ei = (const int*)d_in[1];       // (2,E): [0..E) src, [E..2E) dst
  const float* ew = (const float*)d_in[2];
  // d_in[3] missing_mask: unused by the reference
  const float* W1 = (const float*)d_in[4];
  const float* b1 = (const float*)d_in[5];
  const float* W2 = (const float*)d_in[6];
  const float* b2 = (const float*)d_in[7];
  float* out = (float*)d_out;

  float* ws   = (float*)d_ws;
  float* deg  = ws + OFF_DEG;    // becomes dis after k_dis
  float* norm = ws + OFF_NORM;
  float* xw   = ws + OFF_XW;     // layer-1 GEMM out; reused as hw for layer 2
  float* h    = ws + OFF_H;

  const long long TN64 = (long long)T_DIM * N_NODES * FH;   // 10,240,000
  const long long TN32 = (long long)T_DIM * N_NODES * FIN;  //  5,120,000

  // normalization coefficients
  k_init_deg <<<(N_NODES + 255) / 256, 256, 0, stream>>>(deg);
  k_deg_accum<<<E_EDGES / 256,        256, 0, stream>>>(deg, ei, ew);
  k_dis      <<<(N_NODES + 255) / 256, 256, 0, stream>>>(deg);
  k_norm     <<<(M_MSGS + 255) / 256,  256, 0, stream>>>(norm, deg, ei, ew);

  // layer 1: xw = x@W1 ; h = b1 + scatter(norm * xw[src]) ; h = relu(h)
  k_init_bias<<<(unsigned)((TN64 + 255) / 256), 256, 0, stream>>>(h, b1, FH - 1, TN64, TN64);
  k_gemm_wmma<FIN, FH><<<1250, 256, 0, stream>>>(x, W1, xw);
  k_scatter<6><<<dim3((unsigned)((long long)M_MSGS * FH / 256), T_DIM), 256, 0, stream>>>(xw, h, ei, norm);
  k_relu4<<<(unsigned)(TN64 / 4 / 256), 256, 0, stream>>>((float4*)h);

  // layer 2: hw = h@W2 ; out = b2 + scatter(norm * hw[src])  (tail incl. tuple scalar -> 0)
  k_gemm_wmma<FH, FIN><<<1250, 256, 0, stream>>>(h, W2, xw);
  k_init_bias<<<(unsigned)(((long long)out_size + 255) / 256), 256, 0, stream>>>(out, b2, FIN - 1, TN32, (long long)out_size);
  k_scatter<5><<<dim3((unsigned)((long long)M_MSGS * FIN / 256), T_DIM), 256, 0, stream>>>(xw, out, ei, norm);
}